// GCN_lw_23931557773764
// MI455X (gfx1250) — compile-verified
//
#include <hip/hip_runtime.h>
#include <hip/hip_bf16.h>

typedef float v2f __attribute__((ext_vector_type(2)));
typedef float v8f __attribute__((ext_vector_type(8)));

#define IN_F  128
#define HID_F 64

// ---------------------------------------------------------------------------
// WMMA f32 GEMM: C[nrows x 64] = act(A[nrows x K]) @ B[K x 64]
// One wave -> 16 rows x 64 cols (4 WMMA column tiles), K stepped by 4 using
// V_WMMA_F32_16X16X4_F32. B is staged in LDS (K*64 floats <= 32KB).
//
// f32 A 16x4 layout: lane L holds M=L&15, K=(L>>4)*2 + {0,1} in v[0],v[1].
// f32 B 4x16 layout: lane L holds N=L&15, K=(L>>4)*2 + {0,1} in v[0],v[1].
// f32 C 16x16 layout: lane L holds N=L&15, VGPR r -> M = r + 8*(L>>4).
// ---------------------------------------------------------------------------
template <int K, bool RELU>
__global__ __launch_bounds__(256)
void gemm_wmma_f32(const float* __restrict__ A, const float* __restrict__ B,
                   float* __restrict__ C, int nrows) {
  __shared__ float Bs[K * 64];

  // Cooperative stage of the (tiny) weight matrix into LDS.
  for (int i = threadIdx.x; i < K * 64; i += 256) Bs[i] = B[i];
  __syncthreads();

  const int wave = threadIdx.x >> 5;     // 8 waves / block
  const int lane = threadIdx.x & 31;
  const int row0 = (blockIdx.x * 8 + wave) * 16;
  if (row0 >= nrows) return;             // wave-uniform: EXEC stays all-ones

  const int m    = lane & 15;            // A row within tile / B-D column
  const int kb   = (lane >> 4) << 1;     // 0 or 2: lane-half K offset
  const int half = lane >> 4;

  int arow = row0 + m;
  if (arow >= nrows) arow = nrows - 1;   // safety clamp (N % 16 == 0 normally)
  const float* __restrict__ ap = A + (size_t)arow * K;

  v8f acc0 = {}, acc1 = {}, acc2 = {}, acc3 = {};

  for (int k = 0; k < K; k += 4) {
    float a0 = ap[k + kb];
    float a1 = ap[k + kb + 1];
    if (RELU) { a0 = fmaxf(a0, 0.0f); a1 = fmaxf(a1, 0.0f); }
    v2f a; a.x = a0; a.y = a1;

    const float* brow0 = &Bs[(k + kb) * 64 + m];
    const float* brow1 = &Bs[(k + kb + 1) * 64 + m];

    v2f b;
    b.x = brow0[0];  b.y = brow1[0];
    acc0 = __builtin_amdgcn_wmma_f32_16x16x4_f32(false, a, false, b, (short)0, acc0, false, false);
    b.x = brow0[16]; b.y = brow1[16];
    acc1 = __builtin_amdgcn_wmma_f32_16x16x4_f32(false, a, false, b, (short)0, acc1, false, false);
    b.x = brow0[32]; b.y = brow1[32];
    acc2 = __builtin_amdgcn_wmma_f32_16x16x4_f32(false, a, false, b, (short)0, acc2, false, false);
    b.x = brow0[48]; b.y = brow1[48];
    acc3 = __builtin_amdgcn_wmma_f32_16x16x4_f32(false, a, false, b, (short)0, acc3, false, false);
  }

  // Store D tiles: row = row0 + r + 8*half, col = 16*t + m.
#pragma unroll
  for (int r = 0; r < 8; ++r) {
    const int row = row0 + r + 8 * half;
    if (row < nrows) {
      float* __restrict__ cp = C + (size_t)row * 64 + m;
      cp[0]  = acc0[r];
      cp[16] = acc1[r];
      cp[32] = acc2[r];
      cp[48] = acc3[r];
    }
  }
}

// ---------------------------------------------------------------------------
// H[n, :] = bias[:]  (broadcast init so the scatter folds the bias in)
// ---------------------------------------------------------------------------
__global__ __launch_bounds__(256)
void init_bias_kernel(float* __restrict__ H, const float* __restrict__ b, int total4) {
  const int tid = blockIdx.x * 256 + threadIdx.x;   // one float4 per thread
  if (tid >= total4) return;
  const int f4 = (tid & 15) << 2;                   // 64 floats = 16 float4 / row
  const float4 v = *reinterpret_cast<const float4*>(b + f4);
  *reinterpret_cast<float4*>(H + (size_t)tid * 4) = v;
}

// ---------------------------------------------------------------------------
// Scatter: H[dst[e], :] += M[src[e], :]. 16 lanes per edge, float4 per lane,
// so each edge's 256B source row is a fully coalesced read followed by
// 4 global_atomic_add_f32 per lane (L2-resident working set).
// ---------------------------------------------------------------------------
__global__ __launch_bounds__(256)
void scatter_add_kernel(const float* __restrict__ M, const int* __restrict__ src,
                        const int* __restrict__ dst, float* __restrict__ H, int E) {
  const int tid = blockIdx.x * 256 + threadIdx.x;
  const int e = tid >> 4;
  if (e >= E) return;
  const int f4 = (tid & 15) << 2;
  const int s = src[e];
  const int d = dst[e];
  const float4 v = *reinterpret_cast<const float4*>(M + (size_t)s * 64 + f4);
  float* __restrict__ out = H + (size_t)d * 64 + f4;
  atomicAdd(out + 0, v.x);
  atomicAdd(out + 1, v.y);
  atomicAdd(out + 2, v.z);
  atomicAdd(out + 3, v.w);
}

// ---------------------------------------------------------------------------
// out = sigmoid(out), in place
// ---------------------------------------------------------------------------
__global__ __launch_bounds__(256)
void sigmoid_kernel(float* __restrict__ O, int total) {
  const int i = blockIdx.x * 256 + threadIdx.x;
  if (i >= total) return;
  const float x = O[i];
  O[i] = 1.0f / (1.0f + __expf(-x));
}

extern "C" void kernel_launch(void* const* d_in, const int* in_sizes, int n_in,
                              void* d_out, int out_size, void* d_ws, size_t ws_size,
                              hipStream_t stream) {
  const float* x   = (const float*)d_in[0];
  const int*   ei  = (const int*)  d_in[1];
  const float* W1  = (const float*)d_in[2];
  const float* b1  = (const float*)d_in[3];
  const float* W2  = (const float*)d_in[4];
  const float* b2  = (const float*)d_in[5];
  const float* W3  = (const float*)d_in[6];
  const float* b3  = (const float*)d_in[7];
  float* out = (float*)d_out;

  const int N = in_sizes[0] / IN_F;
  const int E = in_sizes[1] / 2;
  const int* src = ei;
  const int* dst = ei + E;

  // Ping-pong N x 64 buffers in workspace (2 * 25.6 MB).
  float* bufA = (float*)d_ws;                 // GEMM output (messages)
  float* bufB = bufA + (size_t)N * HID_F;     // scatter accumulator

  const int rowTiles   = (N + 15) / 16;
  const int gemmBlocks = (rowTiles + 7) / 8;          // 8 waves/block
  const int total4     = N * (HID_F / 4);
  const int initBlocks = (total4 + 255) / 256;
  const int scatBlocks = ((E * 16) + 255) / 256;
  const int eltBlocks  = (N * HID_F + 255) / 256;

  // ---- Layer 1: h1 = relu(scatter(x@W1) + b1) ----
  gemm_wmma_f32<IN_F, false><<<gemmBlocks, 256, 0, stream>>>(x, W1, bufA, N);
  init_bias_kernel<<<initBlocks, 256, 0, stream>>>(bufB, b1, total4);
  scatter_add_kernel<<<scatBlocks, 256, 0, stream>>>(bufA, src, dst, bufB, E);
  // relu fused into next GEMM's A load

  // ---- Layer 2: h2 = relu(scatter(relu(h1)@W2) + b2) ----
  gemm_wmma_f32<HID_F, true><<<gemmBlocks, 256, 0, stream>>>(bufB, W2, bufA, N);
  init_bias_kernel<<<initBlocks, 256, 0, stream>>>(bufB, b2, total4);
  scatter_add_kernel<<<scatBlocks, 256, 0, stream>>>(bufA, src, dst, bufB, E);

  // ---- Layer 3: out = sigmoid(scatter(relu(h2)@W3) + b3) ----
  gemm_wmma_f32<HID_F, true><<<gemmBlocks, 256, 0, stream>>>(bufB, W3, bufA, N);
  init_bias_kernel<<<initBlocks, 256, 0, stream>>>(out, b3, total4);
  scatter_add_kernel<<<scatBlocks, 256, 0, stream>>>(bufA, src, dst, out, E);
  sigmoid_kernel<<<eltBlocks, 256, 0, stream>>>(out, N * HID_F);
}